// RFDetModule_64905545777338
// MI455X (gfx1250) — compile-verified
//
#include <hip/hip_runtime.h>
#include <hip/hip_bf16.h>
#include <math.h>

// ---------------- problem constants ----------------
#define BATCH   32
#define H       512
#define W       512
#define NPIX    (H * W)          // 262144 = 2^18
#define NPIX_LOG2 18
#define BORDER  8
#define NMS_K   5
#define NMS_PAD 2
#define TOPK    512
#define GK      15
#define GP      7

#define TILE    32
#define LT      (TILE + 2 * NMS_PAD)   // 36

#define AS1 __attribute__((address_space(1)))
#define AS3 __attribute__((address_space(3)))

// ---------------- CDNA5 async global->LDS copy path ----------------
#if defined(__has_builtin)
#  if __has_builtin(__builtin_amdgcn_global_load_async_to_lds_b32)
#    define HAVE_ASYNC_BUILTIN 1
#  endif
#  if __has_builtin(__builtin_amdgcn_s_wait_asynccnt)
#    define HAVE_WAIT_ASYNC_BUILTIN 1
#  endif
#endif

__device__ __forceinline__ void async_copy_f32_to_lds(float* lds_dst, const float* gsrc) {
#if defined(HAVE_ASYNC_BUILTIN)
    __builtin_amdgcn_global_load_async_to_lds_b32(
        (AS1 int*)gsrc, (AS3 int*)lds_dst, /*offset=*/0, /*cpol=*/0);
#else
    unsigned lds_addr = (unsigned)(unsigned long long)(AS3 void*)lds_dst;
    unsigned long long gaddr = (unsigned long long)gsrc;
    asm volatile("global_load_async_to_lds_b32 %0, %1, off"
                 :: "v"(lds_addr), "v"(gaddr) : "memory");
#endif
}

__device__ __forceinline__ void wait_async_zero() {
#if defined(HAVE_WAIT_ASYNC_BUILTIN)
    __builtin_amdgcn_s_wait_asynccnt(0);
#else
    asm volatile("s_wait_asynccnt 0x0" ::: "memory");
#endif
}

__device__ __forceinline__ void atomic_add_f32(float* p, float v) {
    // lowers to global_atomic_add_f32 (no CAS loop)
    unsafeAtomicAdd(p, v);
}

// ---------------- K1: border filter + 5x5 NMS -> topk_value ----------------
__global__ __launch_bounds__(256) void k_nms(const float* __restrict__ in,
                                             float* __restrict__ topk_value) {
    __shared__ float tile[LT * LT];
    const int bx = blockIdx.x, by = blockIdx.y, b = blockIdx.z;
    const int t  = threadIdx.x;
    const float* img = in + (size_t)b * NPIX;

    const int ox = bx * TILE - NMS_PAD;
    const int oy = by * TILE - NMS_PAD;

    // Stage tile + halo into LDS. Interior pixels (surviving the border
    // filter) go through the async global->LDS data mover; everything else
    // (image edge, border frame) is a zero after border filtering.
    for (int i = t; i < LT * LT; i += 256) {
        const int ly = i / LT, lx = i % LT;
        const int gy = oy + ly, gx = ox + lx;
        const bool interior = (gy >= BORDER) && (gy < H - BORDER) &&
                              (gx >= BORDER) && (gx < W - BORDER);
        if (interior) {
            async_copy_f32_to_lds(&tile[i], img + (size_t)gy * W + gx);
        } else {
            tile[i] = 0.0f;
        }
    }
    wait_async_zero();
    __syncthreads();

    // 256 threads, 4 pixels each (same column, rows r, r+8, r+16, r+24)
    const int lx = t & (TILE - 1);
    const int r0 = t >> 5;            // 0..7
    #pragma unroll
    for (int j = 0; j < 4; ++j) {
        const int ty = r0 + j * 8;
        const float c = tile[(ty + NMS_PAD) * LT + (lx + NMS_PAD)];
        float m = 0.0f;               // thresholded values are >= 0
        #pragma unroll
        for (int dy = 0; dy < NMS_K; ++dy) {
            #pragma unroll
            for (int dx = 0; dx < NMS_K; ++dx) {
                m = fmaxf(m, tile[(ty + dy) * LT + (lx + dx)]);
            }
        }
        const float cth = fmaxf(c, 0.0f);       // thresh = 0.0
        const float out = (cth >= m) ? c : 0.0f;
        topk_value[(size_t)b * NPIX + (size_t)(by * TILE + ty) * W + (bx * TILE + lx)] = out;
    }
}

// ---------------- K2: exact per-image 512th-largest key (radix select) -----
__global__ __launch_bounds__(1024) void k_select(const float* __restrict__ topk_value,
                                                 unsigned* __restrict__ thr_key,
                                                 unsigned* __restrict__ counts) {
    __shared__ unsigned hist[2048];
    __shared__ unsigned s_prefix, s_kneed;
    const int b = blockIdx.x;
    const int t = threadIdx.x;
    const float* v = topk_value + (size_t)b * NPIX;

    if (t == 0) { s_prefix = 0u; s_kneed = TOPK; }

    // Rounds over float bit-keys (all values >= 0 so uint order == float order):
    //   r0: bits [31:21] (2048 bins)  r1: bits [20:10]  r2: bits [9:0] (1024 bins)
    for (int r = 0; r < 3; ++r) {
        for (int i = t; i < 2048; i += 1024) hist[i] = 0u;
        __syncthreads();
        const unsigned prefix = s_prefix;
        const unsigned kneed  = s_kneed;

        for (int i = t; i < NPIX; i += 1024) {
            const unsigned key = __float_as_uint(v[i]);
            if (r == 0) {
                atomicAdd(&hist[key >> 21], 1u);
            } else if (r == 1) {
                if ((key >> 21) == prefix) atomicAdd(&hist[(key >> 10) & 2047u], 1u);
            } else {
                if ((key >> 10) == prefix) atomicAdd(&hist[key & 1023u], 1u);
            }
        }
        __syncthreads();

        if (t == 0) {
            const int top = (r == 2) ? 1023 : 2047;
            unsigned cum = 0u; int sel = 0;
            for (int i2 = top; i2 >= 0; --i2) {
                const unsigned c = hist[i2];
                if (cum + c >= kneed) { sel = i2; s_kneed = kneed - cum; break; }
                cum += c;
            }
            if (r == 0)      s_prefix = (unsigned)sel;
            else if (r == 1) s_prefix = (prefix << 11) | (unsigned)sel;
            else             s_prefix = (prefix << 10) | (unsigned)sel;  // exact key
        }
        __syncthreads();
    }

    if (t == 0) {
        thr_key[b] = s_prefix;
        counts[b]  = 0u;      // reset point counter for this call (stateless launches)
    }
}

// ---------------- K3: topk_mask, zero x, compact point lists ---------------
__global__ __launch_bounds__(256) void k_mask(const float* __restrict__ topk_value,
                                              const unsigned* __restrict__ thr_key,
                                              float* __restrict__ out_x,
                                              float* __restrict__ out_mask,
                                              unsigned* __restrict__ counts,
                                              unsigned* __restrict__ lists) {
    const unsigned i = blockIdx.x * 256u + threadIdx.x;   // < BATCH*NPIX = 2^23
    const unsigned b   = i >> NPIX_LOG2;
    const unsigned pix = i & (NPIX - 1u);
    const float vv = topk_value[i];
    const bool m = __float_as_uint(vv) >= thr_key[b];
    out_mask[i] = m ? 1.0f : 0.0f;
    out_x[i]    = 0.0f;
    if (m) {
        const unsigned pos = atomicAdd(&counts[b], 1u);
        if (pos < TOPK) lists[b * TOPK + pos] = pix;
    }
}

// ---------------- K4: sparse 15x15 Gaussian scatter ------------------------
__global__ __launch_bounds__(256) void k_scatter(const float* __restrict__ topk_value,
                                                 const unsigned* __restrict__ counts,
                                                 const unsigned* __restrict__ lists,
                                                 float* __restrict__ out_x) {
    const unsigned b = blockIdx.y;
    const unsigned p = blockIdx.x;
    unsigned n = counts[b]; if (n > TOPK) n = TOPK;
    if (p >= n) return;

    const unsigned pix = lists[b * TOPK + p];
    const float val = topk_value[(size_t)b * NPIX + pix];
    if (val == 0.0f) return;

    const int y = (int)(pix >> 9);
    const int x = (int)(pix & 511u);
    const int t = threadIdx.x;
    if (t >= GK * GK) return;
    const int dy = t / GK - GP;
    const int dx = t % GK - GP;
    const int yy = y + dy, xx = x + dx;
    if (yy < 0 || yy >= H || xx < 0 || xx >= W) return;

    // psf[i][j] = exp(-(i-7)^2/(2*0.25)) * exp(-(j-7)^2/(2*0.25))
    const float w = expf(-2.0f * (float)(dy * dy + dx * dx));
    atomic_add_f32(&out_x[(size_t)b * NPIX + (size_t)yy * W + xx], val * w);
}

// ---------------- K5: clip to [0, 1] ---------------------------------------
__global__ __launch_bounds__(256) void k_clip(float* __restrict__ out_x) {
    const unsigned i = blockIdx.x * 256u + threadIdx.x;
    out_x[i] = fminf(fmaxf(out_x[i], 0.0f), 1.0f);
}

// ---------------- host-side launcher ---------------------------------------
extern "C" void kernel_launch(void* const* d_in, const int* in_sizes, int n_in,
                              void* d_out, int out_size, void* d_ws, size_t ws_size,
                              hipStream_t stream) {
    (void)in_sizes; (void)n_in; (void)out_size; (void)ws_size;
    const float* in = (const float*)d_in[0];

    // reference returns (x, topk_mask, topk_value) -> concatenated flat
    float* out_x    = (float*)d_out;
    float* out_mask = out_x + (size_t)BATCH * NPIX;
    float* out_val  = out_x + 2 * (size_t)BATCH * NPIX;

    // workspace layout (all rewritten every call)
    unsigned* thr    = (unsigned*)d_ws;      // 32
    unsigned* counts = thr + 32;             // 32
    unsigned* lists  = counts + 32;          // 32 * 512

    k_nms   <<<dim3(W / TILE, H / TILE, BATCH), 256, 0, stream>>>(in, out_val);
    k_select<<<BATCH, 1024, 0, stream>>>(out_val, thr, counts);
    k_mask  <<<(BATCH * NPIX) / 256, 256, 0, stream>>>(out_val, thr, out_x, out_mask,
                                                       counts, lists);
    k_scatter<<<dim3(TOPK, BATCH), 256, 0, stream>>>(out_val, counts, lists, out_x);
    k_clip  <<<(BATCH * NPIX) / 256, 256, 0, stream>>>(out_x);
}